// TrilinearSampler_34059090658011
// MI455X (gfx1250) — compile-verified
//
#include <hip/hip_runtime.h>

// Trilinear 3D sampling, B=2, H=W=D=128, C=2, N=2^21 points per batch.
// Bandwidth/gather-bound: the 32 MB volume is L2-resident (192 MB L2) and
// gathered with default RT policy; coords/out are streamed once with NT
// policy so they never evict it. wave32: 256-thread blocks = 8 waves,
// 4 points per thread so coords/out move as b128 vectors.
//
// This revision makes the volume base block-uniform (SGPR) so all 32 gathers
// per thread lower to saddr-form global_load_b64 (scalar base + u32 voffset),
// and folds the zero-pad border masks into the per-dimension weights.

typedef float v2f __attribute__((ext_vector_type(2)));
typedef float v4f __attribute__((ext_vector_type(4)));

#define DIM    128
#define HWD    (DIM * DIM * DIM)
#define NPB_LG 21                 // points per batch element = 2^21
#define PADHI  129                // padded dims are 130 -> clip to [0,129]

#define YSTRB  (DIM * DIM * 2 * 4)   // 131072 bytes per +1 in y
#define XSTRB  (DIM * 2 * 4)         // 1024   bytes per +1 in x
#define ZSTRB  (2 * 4)               // 8      bytes per +1 in z

struct DimPrep {
    float w0, w1;   // weights for the i0 / i1 side, pre-masked by in-bounds
    int   o0, o1;   // clamped byte-offset components
};

__device__ __forceinline__ DimPrep prep_dim(float c, int stride_bytes) {
    const float p   = c + 1.0f;            // position in padded space
    const int   i0u = (int)floorf(p);
    const int   i0  = min(max(i0u, 0), PADHI);       // clipped padded indices
    const int   i1  = min(max(i0u + 1, 0), PADHI);   // (matches reference clip)
    const float d   = (float)i1 - p;       // weight paired with the i0 side
    const float e   = 1.0f - d;            // weight paired with the i1 side
    // Real voxel exists iff padded index is in [1,128]; otherwise the padded
    // value is 0, so zero the WEIGHT instead of the value (w*0 == 0*v).
    const bool  m0  = (unsigned)(i0 - 1) < (unsigned)DIM;
    const bool  m1  = (unsigned)(i1 - 1) < (unsigned)DIM;
    const int   j0  = min(max(i0 - 1, 0), DIM - 1);  // always-safe voxel index
    const int   j1  = min(max(i1 - 1, 0), DIM - 1);
    DimPrep r;
    r.w0 = m0 ? d : 0.0f;
    r.w1 = m1 ? e : 0.0f;
    r.o0 = j0 * stride_bytes;
    r.o1 = j1 * stride_bytes;
    return r;
}

__device__ __forceinline__ v2f ld2(const char* __restrict__ imb, unsigned off) {
    return *(const v2f*)(imb + off);       // global_load_b64 saddr + u32 voffset
}

__device__ __forceinline__ v2f sample_point(const char* __restrict__ imb,
                                            float cy, float cx, float cz) {
    const DimPrep Y = prep_dim(cy, YSTRB);
    const DimPrep X = prep_dim(cx, XSTRB);
    const DimPrep Z = prep_dim(cz, ZSTRB);

    const int a00 = Y.o0 + X.o0, a01 = Y.o0 + X.o1;   // (y?,x?) partial offsets
    const int a10 = Y.o1 + X.o0, a11 = Y.o1 + X.o1;

    const v2f v000 = ld2(imb, (unsigned)(a00 + Z.o0));  // (y0,x0,z0)
    const v2f v001 = ld2(imb, (unsigned)(a00 + Z.o1));  // (y0,x0,z1)
    const v2f v010 = ld2(imb, (unsigned)(a01 + Z.o0));  // (y0,x1,z0)
    const v2f v011 = ld2(imb, (unsigned)(a01 + Z.o1));  // (y0,x1,z1)
    const v2f v100 = ld2(imb, (unsigned)(a10 + Z.o0));  // (y1,x0,z0)
    const v2f v101 = ld2(imb, (unsigned)(a10 + Z.o1));  // (y1,x0,z1)
    const v2f v110 = ld2(imb, (unsigned)(a11 + Z.o0));  // (y1,x1,z0)
    const v2f v111 = ld2(imb, (unsigned)(a11 + Z.o1));  // (y1,x1,z1)

    // weight(yA,xB,zC) = Y.wA * X.wB * Z.wC  (dy pairs y0, dx pairs x0, dz pairs z0)
    const float wyx00 = Y.w0 * X.w0, wyx01 = Y.w0 * X.w1;
    const float wyx10 = Y.w1 * X.w0, wyx11 = Y.w1 * X.w1;
    const float w000 = wyx00 * Z.w0, w001 = wyx00 * Z.w1;
    const float w010 = wyx01 * Z.w0, w011 = wyx01 * Z.w1;
    const float w100 = wyx10 * Z.w0, w101 = wyx10 * Z.w1;
    const float w110 = wyx11 * Z.w0, w111 = wyx11 * Z.w1;

    v2f acc;
    acc.x = fmaf(w000, v000.x, fmaf(w001, v001.x, fmaf(w010, v010.x, fmaf(w011, v011.x,
            fmaf(w100, v100.x, fmaf(w101, v101.x, fmaf(w110, v110.x, w111 * v111.x)))))));
    acc.y = fmaf(w000, v000.y, fmaf(w001, v001.y, fmaf(w010, v010.y, fmaf(w011, v011.y,
            fmaf(w100, v100.y, fmaf(w101, v101.y, fmaf(w110, v110.y, w111 * v111.y)))))));
    return acc;
}

__global__ __launch_bounds__(256) void
TrilinearSampler_34059090658011_kernel(const float* __restrict__ im,
                                       const float* __restrict__ coords,
                                       float* __restrict__ out,
                                       long n_groups) {
    const long g = (long)blockIdx.x * blockDim.x + threadIdx.x;  // 4-point group id
    if (g >= n_groups) return;
    const long p0 = g << 2;                       // first point of this group

    // A 256-thread block covers 1024 contiguous points; 2^21 % 1024 == 0, so
    // the batch index is block-uniform -> volume base lives in SGPRs and all
    // gathers use scalar-base + 32-bit voffset (per-batch volume = 16 MB).
    const unsigned b = blockIdx.x >> (NPB_LG - 10);          // blocks of 1024 points
    const char* __restrict__ imb =
        (const char*)im + ((size_t)b << 24);                 // b * 128^3 * 2 * 4 bytes

    // 12 consecutive coord floats for 4 points -> three NT b128 loads.
    const v4f* cp = (const v4f*)(coords + (size_t)p0 * 3);
    const v4f q0 = __builtin_nontemporal_load(cp + 0);
    const v4f q1 = __builtin_nontemporal_load(cp + 1);
    const v4f q2 = __builtin_nontemporal_load(cp + 2);

    // Per-point layout: (y, x, z); reference uses x=coords[...,1], y=[...,0], z=[...,2]
    const v2f r0 = sample_point(imb, q0.x, q0.y, q0.z);
    const v2f r1 = sample_point(imb, q0.w, q1.x, q1.y);
    const v2f r2 = sample_point(imb, q1.z, q1.w, q2.x);
    const v2f r3 = sample_point(imb, q2.y, q2.z, q2.w);

    // 8 output floats -> two NT b128 stores (32*g bytes, 16B aligned).
    v4f o0, o1;
    o0.x = r0.x; o0.y = r0.y; o0.z = r1.x; o0.w = r1.y;
    o1.x = r2.x; o1.y = r2.y; o1.z = r3.x; o1.w = r3.y;
    v4f* op = (v4f*)(out + (size_t)p0 * 2);
    __builtin_nontemporal_store(o0, op + 0);
    __builtin_nontemporal_store(o1, op + 1);
}

extern "C" void kernel_launch(void* const* d_in, const int* in_sizes, int n_in,
                              void* d_out, int out_size, void* d_ws, size_t ws_size,
                              hipStream_t stream) {
    (void)n_in; (void)out_size; (void)d_ws; (void)ws_size;
    const float* im     = (const float*)d_in[0];   // [B,128,128,128,2] f32
    const float* coords = (const float*)d_in[1];   // [B,N,3] f32
    float*       out    = (float*)d_out;           // [B,N,2] f32

    const long n_points = (long)in_sizes[1] / 3;   // B*N = 4,194,304
    const long n_groups = n_points >> 2;           // 4 points per thread
    const int  block    = 256;                      // 8 wave32 waves
    const long grid     = (n_groups + block - 1) / block;

    TrilinearSampler_34059090658011_kernel<<<(int)grid, block, 0, stream>>>(
        im, coords, out, n_groups);
}